// GNN_layer_SIM_8589934592121
// MI455X (gfx1250) — compile-verified
//
#include <hip/hip_runtime.h>
#include <math.h>
#include <stdint.h>

#define NN 2048
#define LL 3072
#define EE 32768
#define OC 8

typedef __attribute__((ext_vector_type(2))) float v2f;
typedef __attribute__((ext_vector_type(8))) float v8f;

// ---------------------------------------------------------------------------
// Kernel 0: integer shift amounts sp/ss = int(dt * L)  (truncation, dt >= 0)
// ---------------------------------------------------------------------------
__global__ void k_shifts(const float* __restrict__ dtp, const float* __restrict__ dts,
                         int* __restrict__ shp, int* __restrict__ shs) {
  int n = blockIdx.x * blockDim.x + threadIdx.x;
  if (n < NN) {
    shp[n] = (int)(dtp[n] * (float)LL);
    shs[n] = (int)(dts[n] * (float)LL);
  }
}

// ---------------------------------------------------------------------------
// Kernel 1: fused Conv1d(k=7)+BN+ReLU x2 via V_WMMA_F32_16X16X4_F32.
// One block = one node x 1024 positions. x tile staged to LDS with
// GLOBAL_LOAD_ASYNC_TO_LDS_B32 (ASYNCcnt). A 4th all-zero x row absorbs the
// K-padding (21->24) so the WMMA inner loops are branch-free.
// Also emits pristine copies of channels 0/1 into ws (for the graph stage).
// ---------------------------------------------------------------------------
__global__ void __launch_bounds__(256)
k_conv(const float* __restrict__ x,
       const float* __restrict__ w1, const float* __restrict__ b1,
       const float* __restrict__ g1, const float* __restrict__ be1,
       const float* __restrict__ m1, const float* __restrict__ v1p,
       const float* __restrict__ w2, const float* __restrict__ b2,
       const float* __restrict__ g2, const float* __restrict__ be2,
       const float* __restrict__ m2, const float* __restrict__ v2p,
       float* __restrict__ out, float* __restrict__ vch0, float* __restrict__ vch1)
{
  __shared__ float xs[4][1048];      // x tile [p0-6, p0+1040); row 3 = zeros
  __shared__ float v1s[8][1044];     // conv1 out, positions [p0-3, p0+1037)
  __shared__ float sc1[8], bi1[8], sc2[8], bi2[8];
  __shared__ float w1s[8 * 24];      // K padded 21 -> 24 (zeros)
  __shared__ float w2s[8 * 56];

  const int n   = blockIdx.x / 3;
  const int p0  = (blockIdx.x % 3) * 1024;
  const int tid = threadIdx.x;

  if (tid < 8) {
    float s1 = g1[tid] * rsqrtf(v1p[tid] + 1e-5f);
    sc1[tid] = s1;
    bi1[tid] = (b1[tid] - m1[tid]) * s1 + be1[tid];
    float s2 = g2[tid] * rsqrtf(v2p[tid] + 1e-5f);
    sc2[tid] = s2;
    bi2[tid] = (b2[tid] - m2[tid]) * s2 + be2[tid];
  }
  for (int i = tid; i < 8 * 24; i += 256) {
    int k = i % 24;
    w1s[i] = (k < 21) ? w1[(i / 24) * 21 + k] : 0.0f;
  }
  for (int i = tid; i < 8 * 56; i += 256) w2s[i] = w2[i];

  // ---- async DMA of the x tile into LDS (per-lane global->LDS, ASYNCcnt) ----
  const float* xrow = x + (size_t)n * (3 * LL);
  uint32_t xs_lo = (uint32_t)(uintptr_t)(&xs[0][0]);   // LDS byte address
  for (int i = tid; i < 3 * 1046; i += 256) {
    int ci = i / 1046, idx = i - ci * 1046;
    int gp = p0 - 6 + idx;
    if (gp >= 0 && gp < LL) {
      uint64_t ga = (uint64_t)(uintptr_t)(xrow + ci * LL + gp);
      uint32_t la = xs_lo + (uint32_t)(ci * 1048 + idx) * 4u;
      asm volatile("global_load_async_to_lds_b32 %0, %1, off"
                   :: "v"(la), "v"(ga) : "memory");
    } else {
      xs[ci][idx] = 0.0f;              // conv1 zero-padding at sequence edges
    }
  }
  for (int i = tid; i < 1048; i += 256) xs[3][i] = 0.0f;   // K-pad row
  asm volatile("s_wait_asynccnt 0x0" ::: "memory");
  __syncthreads();

  const int wave = tid >> 5;
  const int lane = tid & 31;
  const int lcol = lane & 15;
  const int hi   = lane >> 4;          // 0: lanes 0-15 (K+0/1), 1: lanes 16-31 (K+2/3)

  // ---- conv1: 65 tiles cover positions [p0-3, p0+1037), K = 24 (6 steps) ----
  v2f a1[6];
  #pragma unroll
  for (int s = 0; s < 6; ++s) {
    int kk = 4 * s + 2 * hi;           // padded weights already zero for kk >= 21
    a1[s].x = (lcol < 8) ? w1s[lcol * 24 + kk]     : 0.0f;
    a1[s].y = (lcol < 8) ? w1s[lcol * 24 + kk + 1] : 0.0f;
  }
  for (int t = wave; t < 65; t += 8) {
    int base = 16 * t;
    v8f c = {};
    #pragma unroll
    for (int s = 0; s < 6; ++s) {
      int kk  = 4 * s + 2 * hi;        // kk in [0,23]; ci=3 hits the zero row
      int ci0 = kk / 7,       dk0 = kk - ci0 * 7;
      int ci1 = (kk + 1) / 7, dk1 = (kk + 1) - ci1 * 7;
      v2f b;
      b.x = xs[ci0][base + lcol + dk0];
      b.y = xs[ci1][base + lcol + dk1];
      c = __builtin_amdgcn_wmma_f32_16x16x4_f32(false, a1[s], false, b,
                                                (short)0, c, false, false);
    }
    if (hi == 0) {                     // rows 0..7 = real channels
      int pos = base + lcol;
      int gp  = p0 - 3 + pos;
      int inb = (gp >= 0 && gp < LL);  // conv2 zero-padding at sequence edges
      #pragma unroll
      for (int r = 0; r < 8; ++r)
        v1s[r][pos] = inb ? fmaxf(sc1[r] * c[r] + bi1[r], 0.0f) : 0.0f;
    }
  }
  __syncthreads();

  // ---- conv2: 64 tiles cover positions [p0, p0+1024), K = 56 (14 steps) ----
  v2f a2[14];
  #pragma unroll
  for (int s = 0; s < 14; ++s) {
    int kk = 4 * s + 2 * hi;
    a2[s].x = (lcol < 8) ? w2s[lcol * 56 + kk]     : 0.0f;
    a2[s].y = (lcol < 8) ? w2s[lcol * 56 + kk + 1] : 0.0f;
  }
  float* orow = out + (size_t)n * (OC * LL);
  for (int t = wave; t < 64; t += 8) {
    int base = 16 * t;
    v8f c = {};
    #pragma unroll
    for (int s = 0; s < 14; ++s) {
      int kk  = 4 * s + 2 * hi;        // kk in [0,55], always valid
      int ci0 = kk / 7,       dk0 = kk - ci0 * 7;
      int ci1 = (kk + 1) / 7, dk1 = (kk + 1) - ci1 * 7;
      v2f b;
      b.x = v1s[ci0][base + lcol + dk0];
      b.y = v1s[ci1][base + lcol + dk1];
      c = __builtin_amdgcn_wmma_f32_16x16x4_f32(false, a2[s], false, b,
                                                (short)0, c, false, false);
    }
    if (hi == 0) {
      int gp = p0 + base + lcol;
      #pragma unroll
      for (int r = 0; r < 8; ++r) {
        float val = fmaxf(sc2[r] * c[r] + bi2[r], 0.0f);
        orow[r * LL + gp] = val;
        if (r == 0) vch0[(size_t)n * LL + gp] = val;  // pristine channel copies
        if (r == 1) vch1[(size_t)n * LL + gp] = val;
      }
    }
  }
}

// ---------------------------------------------------------------------------
// CSR build over destinations (channel-independent, built once per call).
// ---------------------------------------------------------------------------
__global__ void k_zero_csr(int* __restrict__ degi, int* __restrict__ cur) {
  int i = blockIdx.x * blockDim.x + threadIdx.x;
  if (i < NN) { degi[i] = 0; cur[i] = 0; }
}

__global__ void k_count(const int* __restrict__ ei, int* __restrict__ degi) {
  int e = blockIdx.x * blockDim.x + threadIdx.x;
  if (e < EE) atomicAdd(&degi[ei[EE + e]], 1);
}

__global__ void __launch_bounds__(256)
k_scan(const int* __restrict__ degi, int* __restrict__ off) {
  __shared__ int ps[256];
  int t = threadIdx.x;
  int base = t * 8;
  int loc[8];
  int sum = 0;
  #pragma unroll
  for (int k = 0; k < 8; ++k) { loc[k] = sum; sum += degi[base + k]; }
  ps[t] = sum;
  __syncthreads();
  for (int d = 1; d < 256; d <<= 1) {   // inclusive scan of chunk sums
    int v = (t >= d) ? ps[t - d] : 0;
    __syncthreads();
    ps[t] += v;
    __syncthreads();
  }
  int prev = (t == 0) ? 0 : ps[t - 1];
  #pragma unroll
  for (int k = 0; k < 8; ++k) off[base + k] = prev + loc[k];
  if (t == 255) off[NN] = prev + sum;
}

__global__ void k_fill(const int* __restrict__ ei, const int* __restrict__ off,
                       int* __restrict__ cur, int* __restrict__ elist) {
  int e = blockIdx.x * blockDim.x + threadIdx.x;
  if (e >= EE) return;
  int d = ei[EE + e];
  int slot = atomicAdd(&cur[d], 1);
  elist[off[d] + slot] = e;
}

// ---------------------------------------------------------------------------
// Per-channel graph stage. Rolls folded into indexing:
//   t[n][k] = value[n][(k - sh[n]) mod L]
// ---------------------------------------------------------------------------
__global__ void k_zero_amax(unsigned* __restrict__ amax) {
  int i = blockIdx.x * blockDim.x + threadIdx.x;
  if (i < NN) amax[i] = 0u;     // 0u == +0.0f; alpha > 0 so uint-max works
}

__global__ void k_alpha(const float* __restrict__ vch, const int* __restrict__ ei,
                        const int* __restrict__ sh, float* __restrict__ alpha,
                        unsigned* __restrict__ amax, int w0) {
  int e = blockIdx.x * blockDim.x + threadIdx.x;
  if (e >= EE) return;
  int src = ei[e], dst = ei[EE + e];
  int sd = sh[dst], so = sh[src];
  const float* vd = vch + (size_t)dst * LL;
  const float* vs = vch + (size_t)src * LL;
  float sim = 0.0f;
  #pragma unroll
  for (int k = 0; k < 20; ++k) {
    int kk = w0 + k;
    int id = kk - sd; if (id < 0) id += LL;
    int is = kk - so; if (is < 0) is += LL;
    sim += fabsf(vd[id] - vs[is]);
  }
  float a = expf(-sim);
  alpha[e] = a;
  atomicMax(&amax[dst], __float_as_uint(a));
}

__global__ void k_ew(const int* __restrict__ ei, const float* __restrict__ alpha,
                     const unsigned* __restrict__ amax, float* __restrict__ ew) {
  int e = blockIdx.x * blockDim.x + threadIdx.x;
  if (e >= EE) return;
  ew[e] = expf(alpha[e] - __uint_as_float(amax[ei[EE + e]]));
}

// One block per destination node: accumulate all its edges' messages in
// registers (no output atomics), single out += write. Gathers stay in L2.
__global__ void __launch_bounds__(256)
k_aggr(const float* __restrict__ vch, const int* __restrict__ ei,
       const int* __restrict__ sh, const float* __restrict__ ew,
       const int* __restrict__ off, const int* __restrict__ elist,
       float* __restrict__ out, int c) {
  int n = blockIdx.x;
  int beg = off[n], end = off[n + 1];
  if (beg == end) return;               // no in-edges: out stays = value
  float den = 0.0f;
  for (int q = beg; q < end; ++q) den += ew[elist[q]];
  float inv = 1.0f / (den * (float)(end - beg));   // 1/(denom * deg)
  int shn = sh[n];
  float acc[12];
  #pragma unroll
  for (int k = 0; k < 12; ++k) acc[k] = 0.0f;
  for (int q = beg; q < end; ++q) {
    int e = elist[q];
    int src = ei[e];
    float coef = ew[e] * inv;
    int delta = shn - sh[src]; if (delta < 0) delta += LL;
    const float* vs = vch + (size_t)src * LL;
    #pragma unroll
    for (int k = 0; k < 12; ++k) {
      int i = k * 256 + threadIdx.x;
      int j = i + delta; if (j >= LL) j -= LL;
      acc[k] += coef * vs[j];
    }
  }
  float* od = out + ((size_t)n * OC + c) * LL;
  #pragma unroll
  for (int k = 0; k < 12; ++k) {
    int i = k * 256 + threadIdx.x;
    od[i] += acc[k];                    // residual: out already holds value
  }
}

// ---------------------------------------------------------------------------
extern "C" void kernel_launch(void* const* d_in, const int* in_sizes, int n_in,
                              void* d_out, int out_size, void* d_ws, size_t ws_size,
                              hipStream_t stream) {
  (void)in_sizes; (void)n_in; (void)out_size; (void)ws_size;
  const float* x   = (const float*)d_in[0];
  const int*   ei  = (const int*)  d_in[1];
  const float* dtp = (const float*)d_in[2];
  const float* dts = (const float*)d_in[3];
  const float* w1  = (const float*)d_in[4];
  const float* b1  = (const float*)d_in[5];
  const float* g1  = (const float*)d_in[6];
  const float* be1 = (const float*)d_in[7];
  const float* m1  = (const float*)d_in[8];
  const float* v1  = (const float*)d_in[9];
  const float* w2  = (const float*)d_in[10];
  const float* b2  = (const float*)d_in[11];
  const float* g2  = (const float*)d_in[12];
  const float* be2 = (const float*)d_in[13];
  const float* m2  = (const float*)d_in[14];
  const float* v2  = (const float*)d_in[15];
  float* out = (float*)d_out;

  char* ws = (char*)d_ws;
  int*      shp   = (int*)ws;       ws += (size_t)NN * 4;
  int*      shs   = (int*)ws;       ws += (size_t)NN * 4;
  int*      degi  = (int*)ws;       ws += (size_t)NN * 4;
  int*      cur   = (int*)ws;       ws += (size_t)NN * 4;
  int*      off   = (int*)ws;       ws += (size_t)(NN + 1) * 4;
  unsigned* amax  = (unsigned*)ws;  ws += (size_t)NN * 4;
  int*      elist = (int*)ws;       ws += (size_t)EE * 4;
  float*    alpha = (float*)ws;     ws += (size_t)EE * 4;
  float*    ew    = (float*)ws;     ws += (size_t)EE * 4;
  float*    vch0  = (float*)ws;     ws += (size_t)NN * LL * 4;
  float*    vch1  = (float*)ws;     ws += (size_t)NN * LL * 4;

  k_shifts<<<(NN + 255) / 256, 256, 0, stream>>>(dtp, dts, shp, shs);

  k_conv<<<NN * (LL / 1024), 256, 0, stream>>>(x, w1, b1, g1, be1, m1, v1,
                                               w2, b2, g2, be2, m2, v2,
                                               out, vch0, vch1);

  // CSR by destination (channel independent)
  k_zero_csr<<<(NN + 255) / 256, 256, 0, stream>>>(degi, cur);
  k_count<<<(EE + 255) / 256, 256, 0, stream>>>(ei, degi);
  k_scan<<<1, 256, 0, stream>>>(degi, off);
  k_fill<<<(EE + 255) / 256, 256, 0, stream>>>(ei, off, cur, elist);

  for (int c = 0; c < 2; ++c) {
    const float* vc = c ? vch1 : vch0;
    const int*   sh = c ? shs : shp;
    int w0 = c ? 1490 : 990;
    k_zero_amax<<<(NN + 255) / 256, 256, 0, stream>>>(amax);
    k_alpha<<<(EE + 255) / 256, 256, 0, stream>>>(vc, ei, sh, alpha, amax, w0);
    k_ew<<<(EE + 255) / 256, 256, 0, stream>>>(ei, alpha, amax, ew);
    k_aggr<<<NN, 256, 0, stream>>>(vc, ei, sh, ew, off, elist, out, c);
  }
}